// UnifiedGATExpert_28948079575200
// MI455X (gfx1250) — compile-verified
//
#include <hip/hip_runtime.h>
#include <hip/hip_bf16.h>
#include <math.h>

typedef __attribute__((ext_vector_type(2))) float v2f;
typedef __attribute__((ext_vector_type(8))) float v8f;

#define NN 20000
#define EE 320000
#define ET (EE + NN)
#define HH 64

// ---------- helpers ----------
__device__ __forceinline__ unsigned fenc(float f) {
    unsigned u = __float_as_uint(f);
    return (u & 0x80000000u) ? ~u : (u | 0x80000000u);
}
__device__ __forceinline__ float fdec(unsigned u) {
    return (u & 0x80000000u) ? __uint_as_float(u & 0x7FFFFFFFu) : __uint_as_float(~u);
}
__device__ __forceinline__ void edge_sd(const int* __restrict__ ei, int e, int& s, int& d) {
    if (e < EE) { s = ei[e]; d = ei[EE + e]; } else { s = d = e - EE; }
}

// ---------- tiny: prompt = mean(sp_emb rows) ----------
__global__ void prompt_kernel(const float* __restrict__ sp_emb, float* __restrict__ prompt) {
    int j = threadIdx.x;
    float s = 0.f;
    #pragma unroll
    for (int p = 0; p < 8; ++p) s += sp_emb[p * HH + j];
    prompt[j] = s * 0.125f;
}

// ---------- WMMA fp32 GEMM: out[n, j] = act(sum_k A[n,k] * W[j,k] + bias[j]) ----------
// K = 64 fixed. Block = 128 thr (4 waves); block computes 16 rows x 64 cols.
__global__ __launch_bounds__(128)
void gemm_nt_wmma(const float* __restrict__ A, const float* __restrict__ W,
                  const float* __restrict__ bias, float* __restrict__ out,
                  int nout_total, int relu) {
    __shared__ float As[16][68];
    __shared__ float Ws[64][68];
    const int tid = threadIdx.x;
    const int rowbase = blockIdx.x * 16;
    const int jbase = blockIdx.y * 64;

    const float* Ab = A + (size_t)rowbase * HH;
    #pragma unroll
    for (int i = tid; i < 16 * 64; i += 128) As[i >> 6][i & 63] = Ab[i];
    const float* Wb = W + (size_t)jbase * HH;
    #pragma unroll
    for (int i = tid; i < 64 * 64; i += 128) Ws[i >> 6][i & 63] = Wb[i];
    __syncthreads();

    const int lane = tid & 31;
    const int wave = tid >> 5;
    const int mrow = lane & 15;              // M row (A) / N col (B) within tile
    const int koff = (lane >> 4) << 1;       // lanes 16-31 hold K=2,3
    const int jl   = (wave << 4) + mrow;     // local W row for B fragment

    v8f acc = {};
    #pragma unroll
    for (int kk = 0; kk < 16; ++kk) {
        const int kb = kk * 4 + koff;
        v2f a, b;
        a[0] = As[mrow][kb]; a[1] = As[mrow][kb + 1];
        b[0] = Ws[jl][kb];   b[1] = Ws[jl][kb + 1];
        acc = __builtin_amdgcn_wmma_f32_16x16x4_f32(false, a, false, b,
                                                    (short)0, acc, false, false);
    }

    const int col = jbase + (wave << 4) + mrow;
    const float bv = bias ? bias[col] : 0.0f;
    const int rofs = (lane >> 4) << 3;       // lanes 16-31 -> M+8
    #pragma unroll
    for (int r = 0; r < 8; ++r) {
        float v = acc[r] + bv;
        if (relu) v = fmaxf(v, 0.0f);
        out[(size_t)(rowbase + rofs + r) * nout_total + col] = v;
    }
}

// ---------- xp = LN((1-fw)*spec + fw*prompt) ; one wave per node ----------
__global__ __launch_bounds__(256)
void softprompt_ln_kernel(const float* __restrict__ spec, const float* __restrict__ prompt,
                          const float* __restrict__ sp_fw, const float* __restrict__ lnw,
                          const float* __restrict__ lnb, float* __restrict__ xp) {
    int node = blockIdx.x * 8 + (threadIdx.x >> 5);
    int lane = threadIdx.x & 31;
    if (node >= NN) return;
    float fw = 1.0f / (1.0f + expf(-sp_fw[0]));
    float v0 = (1.0f - fw) * spec[(size_t)node * HH + lane]      + fw * prompt[lane];
    float v1 = (1.0f - fw) * spec[(size_t)node * HH + lane + 32] + fw * prompt[lane + 32];
    float s = v0 + v1;
    for (int o = 16; o; o >>= 1) s += __shfl_xor(s, o, 32);
    float mu = s * (1.0f / 64.0f);
    float d0 = v0 - mu, d1 = v1 - mu;
    float q = d0 * d0 + d1 * d1;
    for (int o = 16; o; o >>= 1) q += __shfl_xor(q, o, 32);
    float r = rsqrtf(q * (1.0f / 64.0f) + 1e-5f);
    xp[(size_t)node * HH + lane]      = d0 * r * lnw[lane]      + lnb[lane];
    xp[(size_t)node * HH + lane + 32] = d1 * r * lnw[lane + 32] + lnb[lane + 32];
}

// ---------- attention alpha terms: al_s[n,h]=<xh[n,h,:],a_s[h,:]>, same for a_d ----------
__global__ __launch_bounds__(256)
void attn_logits_kernel(const float* __restrict__ xh, const float* __restrict__ a_s,
                        const float* __restrict__ a_d, float* __restrict__ al_s,
                        float* __restrict__ al_d, int heads) {
    int node = blockIdx.x * 8 + (threadIdx.x >> 5);
    int lane = threadIdx.x & 31;
    if (node >= NN) return;
    int D = heads * 64;
    int c = D >> 5;                 // elems per lane: 8 (heads=4) or 2 (heads=1)
    int base = lane * c;
    const float* xr = xh + (size_t)node * D;
    float ss = 0.f, dd = 0.f;
    for (int i = 0; i < c; ++i) {
        float xv = xr[base + i];
        ss += xv * a_s[base + i];
        dd += xv * a_d[base + i];
    }
    int lph = 32 / heads;           // lanes per head: 8 or 32
    for (int o = lph >> 1; o; o >>= 1) {
        ss += __shfl_xor(ss, o, 32);
        dd += __shfl_xor(dd, o, 32);
    }
    if ((lane & (lph - 1)) == 0) {
        int h = lane / lph;
        al_s[node * heads + h] = ss;
        al_d[node * heads + h] = dd;
    }
}

// ---------- edge pass A: logit + segment max (encoded uint atomicMax) ----------
__global__ void edge_logit_max_kernel(const int* __restrict__ ei, const float* __restrict__ al_s,
                                      const float* __restrict__ al_d, float* __restrict__ ebuf,
                                      unsigned* __restrict__ mbuf, int heads, int total) {
    int t = blockIdx.x * blockDim.x + threadIdx.x;
    if (t >= total) return;
    int e = t / heads, h = t - e * heads;
    int s, d; edge_sd(ei, e, s, d);
    float x = al_s[s * heads + h] + al_d[d * heads + h];
    float l = x >= 0.f ? x : 0.2f * x;     // leaky_relu(0.2)
    ebuf[t] = l;
    atomicMax(&mbuf[d * heads + h], fenc(l));
}

// ---------- edge pass B: e = exp(logit - max); segment sum ----------
__global__ void edge_exp_sum_kernel(const int* __restrict__ ei, const unsigned* __restrict__ mbuf,
                                    float* __restrict__ ebuf, float* __restrict__ sbuf,
                                    int heads, int total) {
    int t = blockIdx.x * blockDim.x + threadIdx.x;
    if (t >= total) return;
    int e = t / heads, h = t - e * heads;
    int s, d; edge_sd(ei, e, s, d);
    float v = expf(ebuf[t] - fdec(mbuf[d * heads + h]));
    ebuf[t] = v;
    atomicAdd(&sbuf[d * heads + h], v);
}

// ---------- edge pass C: agg[dst] += xh[src] * alpha ; one wave per edge ----------
__global__ __launch_bounds__(256)
void edge_agg_kernel(const int* __restrict__ ei, const float* __restrict__ xh,
                     const float* __restrict__ ebuf, const float* __restrict__ sbuf,
                     float* __restrict__ agg, int heads) {
    int widx = (int)((blockIdx.x * (size_t)blockDim.x + threadIdx.x) >> 5);
    int lane = threadIdx.x & 31;
    if (widx >= ET) return;
    int s, d; edge_sd(ei, widx, s, d);
    int D = heads * 64;
    int c = D >> 5;
    int base = lane * c;
    int h = base >> 6;              // all c elems of this lane share one head
    float alpha = ebuf[(size_t)widx * heads + h] / sbuf[d * heads + h];
    const float* xr = xh + (size_t)s * D;
    float* ar = agg + (size_t)d * D;
    for (int i = 0; i < c; ++i) atomicAdd(&ar[base + i], xr[base + i] * alpha);
}

// ---------- h = [relu](LN(mean_heads(agg) + bias [+ res])) ; one wave per node ----------
__global__ __launch_bounds__(256)
void finalize_kernel(const float* __restrict__ agg, const float* __restrict__ bias,
                     const float* __restrict__ res, const float* __restrict__ lnw,
                     const float* __restrict__ lnb, float* __restrict__ out,
                     int heads, int do_ln, int do_relu) {
    int node = blockIdx.x * 8 + (threadIdx.x >> 5);
    int lane = threadIdx.x & 31;
    if (node >= NN) return;
    int D = heads * 64;
    const float* ar = agg + (size_t)node * D;
    float inv = 1.0f / (float)heads;
    float v0 = 0.f, v1 = 0.f;
    for (int h = 0; h < heads; ++h) { v0 += ar[h * 64 + lane]; v1 += ar[h * 64 + lane + 32]; }
    v0 = v0 * inv + bias[lane];
    v1 = v1 * inv + bias[lane + 32];
    if (res) { v0 += res[(size_t)node * HH + lane]; v1 += res[(size_t)node * HH + lane + 32]; }
    if (do_ln) {
        float s = v0 + v1;
        for (int o = 16; o; o >>= 1) s += __shfl_xor(s, o, 32);
        float mu = s * (1.0f / 64.0f);
        float d0 = v0 - mu, d1 = v1 - mu;
        float q = d0 * d0 + d1 * d1;
        for (int o = 16; o; o >>= 1) q += __shfl_xor(q, o, 32);
        float r = rsqrtf(q * (1.0f / 64.0f) + 1e-5f);
        v0 = d0 * r * lnw[lane] + lnb[lane];
        v1 = d1 * r * lnw[lane + 32] + lnb[lane + 32];
    }
    if (do_relu) { v0 = fmaxf(v0, 0.f); v1 = fmaxf(v1, 0.f); }
    out[(size_t)node * HH + lane]      = v0;
    out[(size_t)node * HH + lane + 32] = v1;
}

// ---------- host ----------
extern "C" void kernel_launch(void* const* d_in, const int* in_sizes, int n_in,
                              void* d_out, int out_size, void* d_ws, size_t ws_size,
                              hipStream_t stream) {
    const float* x      = (const float*)d_in[0];
    const int*   ei     = (const int*)d_in[1];
    const float* sp_emb = (const float*)d_in[2];
    const float* sp_w1  = (const float*)d_in[3];
    const float* sp_b1  = (const float*)d_in[4];
    const float* sp_w2  = (const float*)d_in[5];
    const float* sp_b2  = (const float*)d_in[6];
    const float* sp_fw  = (const float*)d_in[7];
    const float* sp_lnw = (const float*)d_in[8];
    const float* sp_lnb = (const float*)d_in[9];
    const float* g1_w   = (const float*)d_in[10];
    const float* g1_as  = (const float*)d_in[11];
    const float* g1_ad  = (const float*)d_in[12];
    const float* g1_b   = (const float*)d_in[13];
    const float* g2_w   = (const float*)d_in[14];
    const float* g2_as  = (const float*)d_in[15];
    const float* g2_ad  = (const float*)d_in[16];
    const float* g2_b   = (const float*)d_in[17];
    const float* g3_w   = (const float*)d_in[18];
    const float* g3_as  = (const float*)d_in[19];
    const float* g3_ad  = (const float*)d_in[20];
    const float* g3_b   = (const float*)d_in[21];
    const float* ln1_w  = (const float*)d_in[22];
    const float* ln1_b  = (const float*)d_in[23];
    const float* ln2_w  = (const float*)d_in[24];
    const float* ln2_b  = (const float*)d_in[25];

    // workspace carve (256B aligned)
    char* p = (char*)d_ws;
    auto carve = [&](size_t bytes) -> void* {
        void* r = (void*)p;
        p += (bytes + 255) & ~(size_t)255;
        return r;
    };
    float*    prompt = (float*)carve(HH * sizeof(float));
    float*    xp     = (float*)carve((size_t)NN * HH * sizeof(float));
    float*    xh     = (float*)carve((size_t)NN * 256 * sizeof(float)); // also hosts t1/spec
    float*    als    = (float*)carve((size_t)NN * 4 * sizeof(float));
    float*    ald    = (float*)carve((size_t)NN * 4 * sizeof(float));
    unsigned* mbuf   = (unsigned*)carve((size_t)NN * 4 * sizeof(unsigned));
    float*    sbuf   = (float*)carve((size_t)NN * 4 * sizeof(float));
    float*    ebuf   = (float*)carve((size_t)ET * 4 * sizeof(float));
    float*    agg    = (float*)carve((size_t)NN * 256 * sizeof(float));
    float*    h1     = (float*)carve((size_t)NN * HH * sizeof(float));
    float*    h2     = (float*)carve((size_t)NN * HH * sizeof(float));
    float*    t1     = xh;             // [N,64]  (alias; dead before GAT layers)
    float*    spec   = xh + (size_t)NN * HH;

    auto run_gat = [&](const float* xin, const float* gw, const float* gas, const float* gad,
                       const float* gb, const float* res, const float* lnw, const float* lnb,
                       int heads, int do_ln, int do_relu, float* hout) {
        int D = heads * 64;
        hipMemsetAsync(mbuf, 0, (size_t)NN * heads * sizeof(unsigned), stream);
        hipMemsetAsync(sbuf, 0, (size_t)NN * heads * sizeof(float), stream);
        hipMemsetAsync(agg, 0, (size_t)NN * D * sizeof(float), stream);
        gemm_nt_wmma<<<dim3(NN / 16, D / 64), 128, 0, stream>>>(xin, gw, nullptr, xh, D, 0);
        attn_logits_kernel<<<NN / 8, 256, 0, stream>>>(xh, gas, gad, als, ald, heads);
        int total = ET * heads;
        int blocks = (total + 255) / 256;
        edge_logit_max_kernel<<<blocks, 256, 0, stream>>>(ei, als, ald, ebuf, mbuf, heads, total);
        edge_exp_sum_kernel<<<blocks, 256, 0, stream>>>(ei, mbuf, ebuf, sbuf, heads, total);
        edge_agg_kernel<<<(ET * 32 + 255) / 256, 256, 0, stream>>>(ei, xh, ebuf, sbuf, agg, heads);
        finalize_kernel<<<NN / 8, 256, 0, stream>>>(agg, gb, res, lnw, lnb, hout,
                                                    heads, do_ln, do_relu);
    };

    // SoftPrompt
    prompt_kernel<<<1, HH, 0, stream>>>(sp_emb, prompt);
    gemm_nt_wmma<<<dim3(NN / 16, 1), 128, 0, stream>>>(x, sp_w1, sp_b1, t1, HH, 1);
    gemm_nt_wmma<<<dim3(NN / 16, 1), 128, 0, stream>>>(t1, sp_w2, sp_b2, spec, HH, 0);
    softprompt_ln_kernel<<<NN / 8, 256, 0, stream>>>(spec, prompt, sp_fw, sp_lnw, sp_lnb, xp);

    // GAT layers
    run_gat(xp, g1_w, g1_as, g1_ad, g1_b, nullptr, ln1_w, ln1_b, 4, 1, 1, h1);
    run_gat(h1, g2_w, g2_as, g2_ad, g2_b, h1, ln2_w, ln2_b, 4, 1, 1, h2);
    run_gat(h2, g3_w, g3_as, g3_ad, g3_b, nullptr, nullptr, nullptr, 1, 0, 0, (float*)d_out);
}